// Radon_54408645705960
// MI455X (gfx1250) — compile-verified
//
#include <hip/hip_runtime.h>
#include <hip/hip_bf16.h>
#include <math.h>

#define IMG_W 512
#define N_THETA 180

typedef __attribute__((ext_vector_type(2))) float v2f;
typedef __attribute__((ext_vector_type(8))) float v8f;

// Branchless bilinear sample with zeros padding (matches reference semantics:
// clamp index for the load, zero the weight when out of bounds).
__device__ __forceinline__ float bilinear_zero(const float* __restrict__ img,
                                               float ix, float iy) {
    float x0f = floorf(ix);
    float y0f = floorf(iy);
    float fx = ix - x0f;
    float fy = iy - y0f;
    int x0 = (int)x0f;
    int y0 = (int)y0f;
    int x1 = x0 + 1;
    int y1 = y0 + 1;

    float vx0 = (x0 >= 0 && x0 < IMG_W) ? 1.0f : 0.0f;
    float vx1 = (x1 >= 0 && x1 < IMG_W) ? 1.0f : 0.0f;
    float vy0 = (y0 >= 0 && y0 < IMG_W) ? 1.0f : 0.0f;
    float vy1 = (y1 >= 0 && y1 < IMG_W) ? 1.0f : 0.0f;

    int cx0 = min(max(x0, 0), IMG_W - 1);
    int cx1 = min(max(x1, 0), IMG_W - 1);
    int cy0 = min(max(y0, 0), IMG_W - 1);
    int cy1 = min(max(y1, 0), IMG_W - 1);

    float p00 = img[cy0 * IMG_W + cx0];
    float p01 = img[cy0 * IMG_W + cx1];
    float p10 = img[cy1 * IMG_W + cx0];
    float p11 = img[cy1 * IMG_W + cx1];

    float gx0 = 1.0f - fx;
    float gy0 = 1.0f - fy;
    float w00 = gx0 * gy0 * vx0 * vy0;
    float w01 = fx  * gy0 * vx1 * vy0;
    float w10 = gx0 * fy  * vx0 * vy1;
    float w11 = fx  * fy  * vx1 * vy1;

    return fmaf(p00, w00, fmaf(p01, w01, fmaf(p10, w10, p11 * w11)));
}

// One block = 256 threads = 8 wave32s. Each wave handles one 16-column tile
// of one (n, theta) projection. The row-sum (einsum with the all-ones/512
// mvec) is performed on the matrix unit: D = A(ones 16x4) * B(4x16 samples)
// + C accumulated over 128 chunks of 4 rows. Every row of D holds the column
// sums; lanes 0..15 of acc[0] carry D[0][n].
__global__ __launch_bounds__(256) void radon_wmma_kernel(
        const float* __restrict__ x, float* __restrict__ out) {
    const int cgroup = blockIdx.x & 3;          // 4 column groups of 128 cols
    const int nt     = blockIdx.x >> 2;         // (n, theta) pair
    const int n      = nt / N_THETA;
    const int t      = nt - n * N_THETA;

    const int wave  = threadIdx.x >> 5;
    const int lane  = threadIdx.x & 31;
    const int ln    = lane & 15;                // column within tile / N index
    const int khalf = (lane >> 4) << 1;         // K rows 0,1 (lanes 0-15) or 2,3

    const float th = (float)t * (3.14159265358979323846f / 180.0f);
    const float st = sinf(th);
    const float ct = cosf(th);

    const int   c  = cgroup * 128 + wave * 16 + ln;
    const float xs = (2.0f * (float)c + 1.0f) * (1.0f / (float)IMG_W) - 1.0f;
    const float yb = (1.0f / (float)IMG_W) - 1.0f;   // ys at r=0

    // ix = ((gx+1)*W - 1)/2 with gx = ct*xs + st*ys(r): affine in r.
    const float ix0 = 256.0f * fmaf(ct, xs, st * yb) + 255.5f;
    const float iy0 = 256.0f * fmaf(-st, xs, ct * yb) + 255.5f;

    const float* __restrict__ img = x + n * (IMG_W * IMG_W);

    v8f acc = {};
    v2f a;
    a.x = 1.0f;                                  // all-ones A: layout-proof
    a.y = 1.0f;

    for (int rb = 0; rb < IMG_W; rb += 4) {
        const float r0 = (float)(rb + khalf);
        const float r1 = r0 + 1.0f;
        v2f b;
        b.x = bilinear_zero(img, fmaf(st, r0, ix0), fmaf(ct, r0, iy0));
        b.y = bilinear_zero(img, fmaf(st, r1, ix0), fmaf(ct, r1, iy0));
        // D = ones(16x4) * B(4x16) + C  ->  every D row = column sums of B
        acc = __builtin_amdgcn_wmma_f32_16x16x4_f32(
            /*neg_a=*/false, a, /*neg_b=*/false, b,
            /*c_mod=*/(short)0, acc, /*reuse_a=*/false, /*reuse_b=*/false);
    }

    if (lane < 16) {
        // out shape (2,1,512,180): out[((n*512)+c)*180 + t]
        out[(n * IMG_W + c) * N_THETA + t] = acc[0] * (1.0f / (float)IMG_W);
    }
}

extern "C" void kernel_launch(void* const* d_in, const int* in_sizes, int n_in,
                              void* d_out, int out_size, void* d_ws, size_t ws_size,
                              hipStream_t stream) {
    (void)in_sizes; (void)n_in; (void)out_size; (void)d_ws; (void)ws_size;
    const float* x   = (const float*)d_in[0];
    float*       out = (float*)d_out;
    // 2 images * 180 angles * 4 column groups = 1440 blocks, 256 threads each.
    dim3 grid(2 * N_THETA * 4);
    dim3 block(256);
    radon_wmma_kernel<<<grid, block, 0, stream>>>(x, out);
}